// GridPooling_3539053052130
// MI455X (gfx1250) — compile-verified
//
#include <hip/hip_runtime.h>
#include <stdint.h>

#define VGRID 32
#define NVOX  (VGRID * VGRID * VGRID)   // 32768
#define FDIM  64

typedef float v4f __attribute__((ext_vector_type(4)));

// ---- order-preserving float <-> uint bijection (handles mixed signs) ----
__device__ __forceinline__ unsigned f2ord(float f) {
    unsigned u = __float_as_uint(f);
    return (u & 0x80000000u) ? ~u : (u | 0x80000000u);
}
__device__ __forceinline__ float ord2f(unsigned o) {
    unsigned u = (o & 0x80000000u) ? (o ^ 0x80000000u) : ~o;
    return __uint_as_float(u);
}

// ws layout: ws[0] = ordered global min (uint), ws[1] = ordered global max (uint)

__global__ void init_ws_kernel(unsigned* __restrict__ ws) {
    if (threadIdx.x == 0) {
        ws[0] = 0xFFFFFFFFu;  // min identity in ordered-uint space
        ws[1] = 0x00000000u;  // max identity
    }
}

// Global min/max over all point coordinates (flat array of n floats).
// b128 vector loads -> wave32 shuffle reduce -> LDS block reduce -> 2 atomics/block.
__global__ void minmax_kernel(const float* __restrict__ pts, int n,
                              unsigned* __restrict__ ws) {
    int tid    = blockIdx.x * blockDim.x + threadIdx.x;
    int stride = gridDim.x * blockDim.x;
    unsigned lmin = 0xFFFFFFFFu, lmax = 0u;

    const v4f* p4 = (const v4f*)pts;
    int n4 = n >> 2;
    for (int i = tid; i < n4; i += stride) {
        v4f v = p4[i];                 // global_load_b128 (RT: re-read by scatter)
        #pragma unroll
        for (int j = 0; j < 4; ++j) {
            unsigned o = f2ord(v[j]);
            lmin = min(lmin, o);
            lmax = max(lmax, o);
        }
    }
    // scalar tail
    for (int i = (n4 << 2) + tid; i < n; i += stride) {
        unsigned o = f2ord(pts[i]);
        lmin = min(lmin, o);
        lmax = max(lmax, o);
    }

    // wave32 reduction (warpSize == 32 on gfx1250)
    #pragma unroll
    for (int off = 16; off > 0; off >>= 1) {
        lmin = min(lmin, (unsigned)__shfl_xor((int)lmin, off, 32));
        lmax = max(lmax, (unsigned)__shfl_xor((int)lmax, off, 32));
    }
    __shared__ unsigned smin[8], smax[8];   // 256 threads -> 8 waves
    int lane = threadIdx.x & 31;
    int wave = threadIdx.x >> 5;
    if (lane == 0) { smin[wave] = lmin; smax[wave] = lmax; }
    __syncthreads();
    if (threadIdx.x == 0) {
        int nw = (blockDim.x + 31) >> 5;
        unsigned bmin = smin[0], bmax = smax[0];
        for (int w = 1; w < nw; ++w) {
            bmin = min(bmin, smin[w]);
            bmax = max(bmax, smax[w]);
        }
        atomicMin(&ws[0], bmin);
        atomicMax(&ws[1], bmax);
    }
}

// Zero-initialize the output grid with b128 stores (regular temporal hint so
// the grid lines are L2-resident before the atomic scatter lands on them).
__global__ void zero_kernel(float4* __restrict__ out, int n4) {
    int i = blockIdx.x * blockDim.x + threadIdx.x;
    if (i < n4) out[i] = make_float4(0.f, 0.f, 0.f, 0.f);
}

// Scatter-max. Grid: (ceil(N*16/256), B). One block = 16 points x 16 chunks.
// Threads 0..15 compute one point's voxel cell each into LDS; all 256 threads
// then read their point's cell from LDS (ds broadcast) -> 16x fewer divides,
// no integer division for the batch index (it's blockIdx.y).
// Positive floats compare identically as ints and the grid is zero-init, so
// only values > 0 need an atomic: global_atomic_max_i32, no return, no CAS.
__global__ void scatter_kernel(const float* __restrict__ pts,
                               const v4f* __restrict__ feat4,
                               const unsigned* __restrict__ ws,
                               int* __restrict__ out,
                               int npts_per_batch) {
    const int b     = blockIdx.y;
    const int pbase = blockIdx.x * 16;                 // first point of this block (in batch)
    const int pin   = pbase + (threadIdx.x >> 4);      // this thread's point (in batch)
    const int c     = threadIdx.x & 15;                // float4 chunk within 64-feat row
    const bool valid = (pin < npts_per_batch);

    __shared__ int cells[16];

    if (threadIdx.x < 16 && (pbase + (int)threadIdx.x) < npts_per_batch) {
        // Uniform scalar loads of the reduced min/max.
        float pmin  = ord2f(ws[0]);
        float pmax  = ord2f(ws[1]);
        float denom = (pmax - pmin) + 1e-6f;

        const int pg = b * npts_per_batch + pbase + threadIdx.x;  // global point
        float x = pts[pg * 3 + 0];
        float y = pts[pg * 3 + 1];
        float z = pts[pg * 3 + 2];
        // Replicate reference f32 math exactly: (v - pmin) / denom, * 32, floor.
        int vx = (int)floorf(((x - pmin) / denom) * 32.0f);
        int vy = (int)floorf(((y - pmin) / denom) * 32.0f);
        int vz = (int)floorf(((z - pmin) / denom) * 32.0f);
        cells[threadIdx.x] = b * NVOX + vx * (VGRID * VGRID) + vy * VGRID + vz;
    }
    __syncthreads();

    if (!valid) return;

    // Coalesced b128 non-temporal load: features are streamed exactly once,
    // keep them from competing with the grid for L2 residency.
    const int fidx = (b * npts_per_batch + pin) * (FDIM / 4) + c;
    v4f v = __builtin_nontemporal_load(&feat4[fidx]);

    const int cell = cells[threadIdx.x >> 4];
    const int base = cell * FDIM + c * 4;

    if (v[0] > 0.f) atomicMax(&out[base + 0], __float_as_int(v[0]));
    if (v[1] > 0.f) atomicMax(&out[base + 1], __float_as_int(v[1]));
    if (v[2] > 0.f) atomicMax(&out[base + 2], __float_as_int(v[2]));
    if (v[3] > 0.f) atomicMax(&out[base + 3], __float_as_int(v[3]));
}

extern "C" void kernel_launch(void* const* d_in, const int* in_sizes, int n_in,
                              void* d_out, int out_size, void* d_ws, size_t ws_size,
                              hipStream_t stream) {
    const float* pts  = (const float*)d_in[0];   // [B, N, 3] f32
    const float* feat = (const float*)d_in[1];   // [B, N, 64] f32
    float*       out  = (float*)d_out;           // [B, 32, 32, 32, 64] f32
    unsigned*    ws   = (unsigned*)d_ws;

    const int nflat = in_sizes[0];               // B*N*3
    const int npts  = nflat / 3;                 // B*N
    const int B     = out_size / (NVOX * FDIM);  // 4
    const int N     = npts / B;                  // 100000

    // 1) reset reduction slots
    init_ws_kernel<<<1, 32, 0, stream>>>(ws);

    // 2) global min/max over all point coords
    minmax_kernel<<<1024, 256, 0, stream>>>(pts, nflat, ws);

    // 3) zero the output grid (b128 stores)
    {
        int n4      = out_size / 4;              // divisible: FDIM==64
        int threads = 256;
        int blocks  = (n4 + threads - 1) / threads;
        zero_kernel<<<blocks, threads, 0, stream>>>((float4*)out, n4);
    }

    // 4) scatter-max via integer atomic-max (positives only)
    {
        int chunks_per_batch = N * (FDIM / 4);   // 16 chunks per point
        dim3 grid((chunks_per_batch + 255) / 256, B);
        scatter_kernel<<<grid, 256, 0, stream>>>(pts, (const v4f*)feat, ws,
                                                 (int*)out, N);
    }
}